// DecoderUnit_19954418057394
// MI455X (gfx1250) — compile-verified
//
#include <hip/hip_runtime.h>
#include <math.h>

#define DD 4096
#define AA 10
#define XPAD 4112   // (D+A)=4106 padded to a multiple of 16 floats (16B async chunks)

typedef float v2f __attribute__((ext_vector_type(2)));
typedef float v4f __attribute__((ext_vector_type(4)));
typedef float v8f __attribute__((ext_vector_type(8)));

// ---------- wave/block reductions (wave32) ----------
__device__ __forceinline__ float warp_sum(float v) {
#pragma unroll
  for (int m = 16; m; m >>= 1) v += __shfl_xor(v, m, 32);
  return v;
}

__device__ __forceinline__ float block_sum(float v, float* red) {
  const int lane = threadIdx.x & 31, wave = threadIdx.x >> 5;
  v = warp_sum(v);
  if (lane == 0) red[wave] = v;
  __syncthreads();
  if (wave == 0) {
    float t = (lane < (int)(blockDim.x >> 5)) ? red[lane] : 0.f;
    t = warp_sum(t);
    if (lane == 0) red[0] = t;
  }
  __syncthreads();
  float r = red[0];
  __syncthreads();
  return r;
}

__device__ __forceinline__ float block_max(float v, float* red) {
  const int lane = threadIdx.x & 31, wave = threadIdx.x >> 5;
#pragma unroll
  for (int m = 16; m; m >>= 1) v = fmaxf(v, __shfl_xor(v, m, 32));
  if (lane == 0) red[wave] = v;
  __syncthreads();
  if (wave == 0) {
    float t = (lane < (int)(blockDim.x >> 5)) ? red[lane] : -INFINITY;
#pragma unroll
    for (int m = 16; m; m >>= 1) t = fmaxf(t, __shfl_xor(t, m, 32));
    if (lane == 0) red[0] = t;
  }
  __syncthreads();
  float r = red[0];
  __syncthreads();
  return r;
}

// ---------- CDNA5 async global->LDS staging of the x vector ----------
__device__ __forceinline__ void stage_x_async(const float* __restrict__ x,
                                              float* xs, int nbytes) {
  unsigned lds_base = (unsigned)(size_t)xs;       // low 32 bits = LDS byte offset
  const char* gp = (const char*)x;
  for (int off = threadIdx.x * 16; off < nbytes; off += blockDim.x * 16) {
    unsigned l = lds_base + (unsigned)off;
    asm volatile("global_load_async_to_lds_b128 %0, %1, off"
                 :: "v"(l), "v"(gp + off) : "memory");
  }
  asm volatile("s_wait_asynccnt 0" ::: "memory");
  __syncthreads();
}

// ---------- softmax(attn) * ctx, concat prev_agent_feat, zero-pad ----------
__global__ __launch_bounds__(1024)
void softmax_concat_kernel(const float* __restrict__ attn, const float* __restrict__ ctx,
                           const float* __restrict__ paf, float* __restrict__ gin) {
  __shared__ float red[32];
  const int tid = threadIdx.x;
  float v[4];
  float mx = -INFINITY;
#pragma unroll
  for (int i = 0; i < 4; ++i) { v[i] = attn[tid + i * 1024]; mx = fmaxf(mx, v[i]); }
  mx = block_max(mx, red);
  float s = 0.f;
#pragma unroll
  for (int i = 0; i < 4; ++i) { v[i] = __expf(v[i] - mx); s += v[i]; }
  s = block_sum(s, red);
  const float inv = 1.f / s;
#pragma unroll
  for (int i = 0; i < 4; ++i) {
    const int idx = tid + i * 1024;
    gin[idx] = v[i] * inv * ctx[idx];
  }
  if (tid < XPAD - DD) gin[DD + tid] = (tid < AA) ? paf[tid] : 0.f;
}

// ---------- GEMV, compile-time COLS % 4 == 0: one row per wave ----------
template <int ACT, int COLS>
__global__ __launch_bounds__(256)
void gemv4_kernel(const float* __restrict__ W, const float* __restrict__ x,
                  const float* __restrict__ b, float* __restrict__ y, int rows) {
  extern __shared__ float xs[];
  stage_x_async(x, xs, COLS * 4);
  const int lane = threadIdx.x & 31;
  const int row = blockIdx.x * 8 + (threadIdx.x >> 5);
  if (row >= rows) return;
  const v4f* __restrict__ wr = (const v4f*)(W + (size_t)row * (size_t)COLS);
  const v4f* __restrict__ xv = (const v4f*)xs;
  constexpr int N4 = COLS >> 2;                 // 1024 -> 32 iters/lane
  float acc = 0.f;
#pragma unroll 8
  for (int j = lane; j < N4; j += 32) {
    v4f w4 = __builtin_nontemporal_load(wr + j);   // NT stream: don't thrash L2
    v4f x4 = xv[j];                                // LDS
    acc = fmaf(w4.x, x4.x, acc);
    acc = fmaf(w4.y, x4.y, acc);
    acc = fmaf(w4.z, x4.z, acc);
    acc = fmaf(w4.w, x4.w, acc);
  }
  acc = warp_sum(acc);
  if (lane == 0) {
    float r = acc + b[row];
    if (ACT) r = fmaxf(r, 0.f);
    y[row] = r;
  }
}

// ---------- GEMV for COLS = 4106 (rows only 8B-aligned): float2 loads ----------
template <int ACT, int COLS, int STAGE_BYTES>
__global__ __launch_bounds__(256)
void gemv2_kernel(const float* __restrict__ W, const float* __restrict__ x,
                  const float* __restrict__ b, float* __restrict__ y, int rows) {
  extern __shared__ float xs[];
  stage_x_async(x, xs, STAGE_BYTES);
  const int lane = threadIdx.x & 31;
  const int row = blockIdx.x * 8 + (threadIdx.x >> 5);
  if (row >= rows) return;
  const v2f* __restrict__ wr = (const v2f*)(W + (size_t)row * (size_t)COLS);
  const v2f* __restrict__ xv = (const v2f*)xs;
  constexpr int N2 = COLS >> 1;                 // 2053
  float acc = 0.f;
#pragma unroll 4
  for (int j = lane; j < N2; j += 32) {
    v2f w2 = __builtin_nontemporal_load(wr + j);
    v2f x2 = xv[j];
    acc = fmaf(w2.x, x2.x, acc);
    acc = fmaf(w2.y, x2.y, acc);
  }
  acc = warp_sum(acc);
  if (lane == 0) {
    float r = acc + b[row];
    if (ACT) r = fmaxf(r, 0.f);
    y[row] = r;
  }
}

// ---------- LayerNorm + ReLU over 4096 (single workgroup) ----------
__global__ __launch_bounds__(1024)
void ln_relu_kernel(const float* __restrict__ x, const float* __restrict__ g,
                    const float* __restrict__ be, float* __restrict__ y) {
  __shared__ float red[32];
  const int tid = threadIdx.x;
  float v[4];
  float s = 0.f;
#pragma unroll
  for (int i = 0; i < 4; ++i) { v[i] = x[tid + i * 1024]; s += v[i]; }
  s = block_sum(s, red);
  const float mu = s * (1.f / DD);
  float q = 0.f;
#pragma unroll
  for (int i = 0; i < 4; ++i) { const float d = v[i] - mu; q = fmaf(d, d, q); }
  q = block_sum(q, red);
  const float rstd = rsqrtf(q * (1.f / DD) + 1e-5f);
#pragma unroll
  for (int i = 0; i < 4; ++i) {
    const int idx = tid + i * 1024;
    const float r = (v[i] - mu) * rstd * g[idx] + be[idx];
    y[idx] = fmaxf(r, 0.f);
  }
}

// ---------- GRU gates ----------
__global__ __launch_bounds__(256)
void gru_gate_kernel(const float* __restrict__ gi, const float* __restrict__ gh,
                     const float* __restrict__ h_prev, float* __restrict__ hid,
                     float* __restrict__ out_hid) {
  const int i = blockIdx.x * blockDim.x + threadIdx.x;
  if (i >= DD) return;
  const float r = 1.f / (1.f + __expf(-(gi[i] + gh[i])));
  const float z = 1.f / (1.f + __expf(-(gi[DD + i] + gh[DD + i])));
  const float n = tanhf(gi[2 * DD + i] + r * gh[2 * DD + i]);
  const float h = (1.f - z) * n + z * h_prev[i];
  hid[i] = h;
  out_hid[i] = h;
}

// ---------- final 10x4096 GEMV via V_WMMA_F32_16X16X4_F32 ----------
// A tile = 16 rows x 4 k of W; rows >= 10 read a clamped-valid row and are
// zeroed by a 0/1 mask multiply (keeps the loop straight-line, no EXEC
// juggling). B = x[k..k+3] broadcast to all 16 N-columns, so every column of
// C accumulates the same dot-product partials. 8 waves split K; partials
// reduced through LDS. Tiny layer (160 KB) vs. the 740 MB streaming floor.
__global__ __launch_bounds__(256)
void gemv_head_wmma(const float* __restrict__ W, const float* __restrict__ x,
                    const float* __restrict__ b, float* __restrict__ y) {
  __shared__ float part[8][16];
  const int lane = threadIdx.x & 31;
  const int wave = threadIdx.x >> 5;   // 8 waves
  const int r = lane & 15;             // A-matrix row held by this lane
  const int khalf = (lane >> 4) * 2;   // lanes 0-15: K=0,1 ; lanes 16-31: K=2,3
  const int rc = (r < AA) ? r : (AA - 1);
  const float msk = (r < AA) ? 1.f : 0.f;
#if __has_builtin(__builtin_amdgcn_wmma_f32_16x16x4_f32)
  v8f c = {};
  const float* __restrict__ wp = W + rc * DD + wave * 512 + khalf;
  const float* __restrict__ xp = x + wave * 512 + khalf;
#pragma unroll 4
  for (int i = 0; i < 128; ++i) {      // K chunk of 4 per WMMA, 512 per wave
    v2f a  = *(const v2f*)wp;
    v2f bb = *(const v2f*)xp;
    a.x *= msk;
    a.y *= msk;
    c = __builtin_amdgcn_wmma_f32_16x16x4_f32(false, a, false, bb,
                                              (short)0, c, false, false);
    wp += 4;
    xp += 4;
  }
  // Column N=0: lane 0 holds M=0..7 (vgpr 0..7), lane 16 holds M=8..15.
  if (lane == 0) {
#pragma unroll
    for (int m = 0; m < 8; ++m) part[wave][m] = c[m];
  } else if (lane == 16) {
#pragma unroll
    for (int m = 0; m < 8; ++m) part[wave][8 + m] = c[m];
  }
#else
  for (int m = 0; m < 16; ++m) {
    float acc = 0.f;
    if (m < AA)
      for (int k = wave * 512 + lane; k < wave * 512 + 512; k += 32)
        acc = fmaf(W[m * DD + k], x[k], acc);
    acc = warp_sum(acc);
    if (lane == 0) part[wave][m] = acc;
  }
#endif
  __syncthreads();
  if (threadIdx.x < AA) {
    float s = 0.f;
#pragma unroll
    for (int w = 0; w < 8; ++w) s += part[w][threadIdx.x];
    y[threadIdx.x] = s + b[threadIdx.x];
  }
}

// ---------- agent_feat head (10 values) ----------
__global__ void head_kernel(const float* __restrict__ o, float* __restrict__ out) {
  if (threadIdx.x == 0) {
    out[0] = o[0];
    out[1] = o[1];
    const float nrm = sqrtf(o[2] * o[2] + o[3] * o[3]);
    out[2] = o[2] / nrm;
    out[3] = o[3] / nrm;
#pragma unroll
    for (int k = 4; k < 7; ++k) {
      const float v = o[k];
      out[k] = (v > 20.f) ? v : log1pf(__expf(v));
    }
    const float m = fmaxf(o[7], fmaxf(o[8], o[9]));
    const float e7 = __expf(o[7] - m), e8 = __expf(o[8] - m), e9 = __expf(o[9] - m);
    const float inv = 1.f / (e7 + e8 + e9);
    out[7] = e7 * inv;
    out[8] = e8 * inv;
    out[9] = e9 * inv;
  }
}

extern "C" void kernel_launch(void* const* d_in, const int* in_sizes, int n_in,
                              void* d_out, int out_size, void* d_ws, size_t ws_size,
                              hipStream_t stream) {
  const float* ctx    = (const float*)d_in[0];
  const float* h_prev = (const float*)d_in[1];
  const float* attn   = (const float*)d_in[2];
  const float* paf    = (const float*)d_in[3];
  const float* inW0 = (const float*)d_in[4];  const float* inb0  = (const float*)d_in[5];
  const float* ing0 = (const float*)d_in[6];  const float* inbe0 = (const float*)d_in[7];
  const float* inW1 = (const float*)d_in[8];  const float* inb1  = (const float*)d_in[9];
  const float* ing1 = (const float*)d_in[10]; const float* inbe1 = (const float*)d_in[11];
  const float* inW2 = (const float*)d_in[12]; const float* inb2  = (const float*)d_in[13];
  const float* Wih  = (const float*)d_in[14]; const float* Whh   = (const float*)d_in[15];
  const float* bih  = (const float*)d_in[16]; const float* bhh   = (const float*)d_in[17];
  const float* oW0  = (const float*)d_in[18]; const float* ob0   = (const float*)d_in[19];
  const float* og0  = (const float*)d_in[20]; const float* obe0  = (const float*)d_in[21];
  const float* oW1  = (const float*)d_in[22]; const float* ob1   = (const float*)d_in[23];
  const float* og1  = (const float*)d_in[24]; const float* obe1  = (const float*)d_in[25];
  const float* oW2  = (const float*)d_in[26]; const float* ob2   = (const float*)d_in[27];
  float* out = (float*)d_out;

  float* ws  = (float*)d_ws;
  float* gin = ws;                // XPAD
  float* t0  = gin + XPAD;        // D
  float* h0  = t0 + DD;           // D
  float* gi  = h0 + DD;           // 3D
  float* gh  = gi + 3 * DD;       // 3D
  float* hid = gh + 3 * DD;       // D
  float* o10 = hid + DD;          // 16

  const dim3 b256(256);
  const size_t smemD = DD * sizeof(float);
  const size_t smemP = XPAD * sizeof(float);

  softmax_concat_kernel<<<1, 1024, 0, stream>>>(attn, ctx, paf, gin);
  gemv2_kernel<0, DD + AA, XPAD * 4><<<DD / 8, b256, smemP, stream>>>(inW0, gin, inb0, t0, DD);
  ln_relu_kernel<<<1, 1024, 0, stream>>>(t0, ing0, inbe0, h0);
  gemv4_kernel<0, DD><<<DD / 8, b256, smemD, stream>>>(inW1, h0, inb1, t0, DD);
  ln_relu_kernel<<<1, 1024, 0, stream>>>(t0, ing1, inbe1, h0);
  gemv4_kernel<1, DD><<<DD / 8, b256, smemD, stream>>>(inW2, h0, inb2, t0, DD); // t0 = relu(gru_in)
  gemv4_kernel<0, DD><<<3 * DD / 8, b256, smemD, stream>>>(Wih, t0, bih, gi, 3 * DD);
  gemv4_kernel<0, DD><<<3 * DD / 8, b256, smemD, stream>>>(Whh, h_prev, bhh, gh, 3 * DD);
  gru_gate_kernel<<<DD / 256, b256, 0, stream>>>(gi, gh, h_prev, hid, out + AA);
  gemv4_kernel<0, DD><<<DD / 8, b256, smemD, stream>>>(oW0, hid, ob0, t0, DD);
  ln_relu_kernel<<<1, 1024, 0, stream>>>(t0, og0, obe0, h0);
  gemv4_kernel<0, DD><<<DD / 8, b256, smemD, stream>>>(oW1, h0, ob1, t0, DD);
  ln_relu_kernel<<<1, 1024, 0, stream>>>(t0, og1, obe1, h0);
  gemv_head_wmma<<<1, b256, 0, stream>>>(oW2, h0, ob2, o10);
  head_kernel<<<1, 32, 0, stream>>>(o10, out);
  (void)in_sizes; (void)n_in; (void)out_size; (void)ws_size;
}